// SoftQuantileChannelAttention_34084860461425
// MI455X (gfx1250) — compile-verified
//
#include <hip/hip_runtime.h>

// ---------------- problem constants (match reference) ----------------
#define B_      16
#define C_      256
#define H_      96
#define W_      128
#define HW_     (H_ * W_)        // 12288
#define OH_     12
#define OW_     16
#define N_      192              // OH_*OW_
#define HIDDEN_ 128
#define Q_      4

typedef __attribute__((ext_vector_type(2))) float v2f;
typedef __attribute__((ext_vector_type(4))) float v4f;
typedef __attribute__((ext_vector_type(8))) float v8f;

// ---------------- kernel 1: 8x8 avg-pool -> flat [B,C,192] ----------------
__global__ __launch_bounds__(256) void k_avgpool(const float* __restrict__ x,
                                                 float* __restrict__ flat) {
    int idx = blockIdx.x * 256 + threadIdx.x;          // < B*C*N
    if (idx >= B_ * C_ * N_) return;
    int n  = idx % N_;
    int bc = idx / N_;
    int ow = n % OW_, oh = n / OW_;
    const v4f* p = (const v4f*)(x + (size_t)bc * HW_ + (size_t)oh * 8 * W_ + ow * 8);
    float s = 0.f;
#pragma unroll
    for (int r = 0; r < 8; ++r) {                      // row stride = W_/4 vec4
        v4f a = __builtin_nontemporal_load(&p[r * (W_ / 4) + 0]);
        v4f b = __builtin_nontemporal_load(&p[r * (W_ / 4) + 1]);
        s += a.x + a.y + a.z + a.w + b.x + b.y + b.z + b.w;
    }
    flat[idx] = s * (1.0f / 64.0f);
}

// ------------- kernel 2: soft ranks + soft quantiles per (b,c) -------------
__global__ __launch_bounds__(256) void k_softquant(const float* __restrict__ flat,
                                                   float* __restrict__ qvals) {
    __shared__ float f[N_];
    __shared__ float red [256];
    __shared__ float red2[256];
    int bc = blockIdx.x;
    int i  = threadIdx.x;
    if (i < N_) f[i] = flat[(size_t)bc * N_ + i];
    __syncthreads();

    float xi = (i < N_) ? f[i] : 0.f;
    float r  = 1.0f;
    if (i < N_) {
        // r_i = 1 + sum_j sigmoid((x_i - x_j)/tau1), 1/tau1 = 10
        for (int j = 0; j < N_; ++j)
            r += 1.0f / (1.0f + __expf((f[j] - xi) * 10.0f));
    }
    const float quant[Q_] = {0.25f, 0.5f, 0.75f, 0.95f};
#pragma unroll
    for (int q = 0; q < Q_; ++q) {
        float target = 1.0f + quant[q] * (float)(N_ - 1);
        float logit  = (i < N_) ? (-fabsf(r - target) * 10.0f) : -1e30f;  // 1/tau2=10
        // block max
        red[i] = logit;
        __syncthreads();
        for (int s = 128; s >= 1; s >>= 1) {
            if (i < s) red[i] = fmaxf(red[i], red[i + s]);
            __syncthreads();
        }
        float mx = red[0];
        __syncthreads();
        // block sum of w and w*f
        float w = (i < N_) ? __expf(logit - mx) : 0.f;
        red [i] = w;
        red2[i] = (i < N_) ? w * f[i] : 0.f;
        __syncthreads();
        for (int s = 128; s >= 1; s >>= 1) {
            if (i < s) { red[i] += red[i + s]; red2[i] += red2[i + s]; }
            __syncthreads();
        }
        if (i == 0) qvals[(size_t)bc * Q_ + q] = red2[0] / red[0];
        __syncthreads();
    }
}

// --------- kernel 3: per-batch MLP -> channel scale = sigmoid(t2) ----------
__global__ __launch_bounds__(256) void k_mlp_scale(const float* __restrict__ qvals,
                                                   const float* __restrict__ w1,
                                                   const float* __restrict__ w2,
                                                   float* __restrict__ scale) {
    __shared__ float qv[C_ * Q_];        // 1024
    __shared__ float t [HIDDEN_ * Q_];   // 512
    int b   = blockIdx.x;
    int tid = threadIdx.x;               // 256 threads
#pragma unroll
    for (int k = 0; k < 4; ++k)
        qv[tid + 256 * k] = qvals[(size_t)b * C_ * Q_ + tid + 256 * k];
    __syncthreads();
#pragma unroll
    for (int k = 0; k < 2; ++k) {
        int idx = tid + 256 * k;         // < 512
        int h = idx >> 2, q = idx & 3;
        float s = 0.f;
        for (int c = 0; c < C_; ++c) s += w1[(size_t)h * C_ + c] * qv[c * Q_ + q];
        t[idx] = fmaxf(s, 0.f);          // relu
    }
    __syncthreads();
    float t2 = 0.f;
    const float* w2r = w2 + (size_t)tid * HIDDEN_ * Q_;   // w2[c][h][q]
    for (int j = 0; j < HIDDEN_ * Q_; ++j) t2 += t[j] * w2r[j];
    scale[(size_t)b * C_ + tid] = 1.0f / (1.0f + __expf(-t2));
}

// ------ kernel 4: Weff[b][c][k] = wf[c,k] + wf[c,C+k] * scale[b,k] --------
__global__ __launch_bounds__(256) void k_weff(const float* __restrict__ wf,
                                              const float* __restrict__ scale,
                                              float* __restrict__ weff) {
    int idx = blockIdx.x * 256 + threadIdx.x;       // < B*C*C
    int k  = idx % C_;
    int bc = idx / C_;
    int c  = bc % C_;
    int b  = bc / C_;
    weff[idx] = wf[(size_t)c * (2 * C_) + k]
              + wf[(size_t)c * (2 * C_) + C_ + k] * scale[(size_t)b * C_ + k];
}

// ------ kernel 5: out[b] = Weff[b] (256x256) @ x[b] (256x12288), WMMA ------
// LDS tile is stored K-quad-interleaved: dword (k,n) -> (k>>2)*(NB*4) + n*4 + (k&3)
// so each WMMA B fragment (two consecutive K at one column) is an adjacent,
// 8-byte-aligned dword pair: one ds_load_b64 straight into a WMMA register
// pair, and a fragment read is bank-conflict-free across all 32 lanes.
#define NTILES 4
#define NB     (NTILES * 16)    // 64 spatial columns per block
__global__ __launch_bounds__(512) void k_fuse_wmma(const float* __restrict__ x,
                                                   const float* __restrict__ weff,
                                                   float* __restrict__ out) {
    __shared__ __align__(16) float xs[C_ * NB];     // 64 KB x-tile
    const int blocksPerBatch = HW_ / NB;            // 192
    int b     = blockIdx.x / blocksPerBatch;
    int ntile = blockIdx.x % blocksPerBatch;
    int n0    = ntile * NB;
    int tid   = threadIdx.x;

    // cooperative load: each task packs 4 K-rows at one column into one b128.
    // consecutive lanes -> consecutive columns -> 128B-contiguous global reads.
    const float* xbase = x + (size_t)b * C_ * HW_ + n0;
#pragma unroll
    for (int it = 0; it < 8; ++it) {
        int i  = tid + 512 * it;                    // < 4096 tasks
        int kq = i >> 6;                            // K quad 0..63
        int n  = i & 63;                            // column 0..63
        const float* gp = xbase + (size_t)(4 * kq) * HW_ + n;
        v4f v;
        v.x = __builtin_nontemporal_load(gp);              // k = 4kq+0
        v.y = __builtin_nontemporal_load(gp +     HW_);    // k = 4kq+1
        v.z = __builtin_nontemporal_load(gp + 2 * HW_);    // k = 4kq+2
        v.w = __builtin_nontemporal_load(gp + 3 * HW_);    // k = 4kq+3
        ((v4f*)xs)[kq * NB + n] = v;
    }
    __syncthreads();

    int wave  = tid >> 5;                           // 16 waves -> 16 M-stripes
    int lane  = tid & 31;
    int hl    = lane >> 4;                          // lane half selects K pair
    int col   = lane & 15;                          // M row (A) / N col (B,C)
    int mbase = wave * 16;

    const float* wrow = weff + ((size_t)b * C_ + (mbase + col)) * C_;

    v8f acc[NTILES];
#pragma unroll
    for (int t = 0; t < NTILES; ++t) acc[t] = (v8f)0.0f;

    for (int kk = 0; kk < C_; kk += 4) {
        // A fragment: 16x4 f32. lane half -> K pair {2hl, 2hl+1}, M = lane%16
        v2f a = *(const v2f*)(wrow + kk + 2 * hl);
        const float* xb = xs + (kk >> 2) * (NB * 4) + 2 * hl;
#pragma unroll
        for (int t = 0; t < NTILES; ++t) {
            // B fragment: adjacent dword pair in interleaved tile
            v2f bb = *(const v2f*)(xb + (t * 16 + col) * 4);
            acc[t] = __builtin_amdgcn_wmma_f32_16x16x4_f32(
                false, a, false, bb, (short)0, acc[t], false, false);
        }
    }

    // C/D layout: VGPR i -> M = mbase + i + 8*hl, N = col; streaming NT stores
    float* ob = out + (size_t)b * C_ * HW_ + n0;
#pragma unroll
    for (int t = 0; t < NTILES; ++t) {
#pragma unroll
        for (int i = 0; i < 8; ++i) {
            int row = mbase + i + hl * 8;
            __builtin_nontemporal_store(acc[t][i],
                                        &ob[(size_t)row * HW_ + t * 16 + col]);
        }
    }
}

// ------------------------------- launcher ---------------------------------
extern "C" void kernel_launch(void* const* d_in, const int* in_sizes, int n_in,
                              void* d_out, int out_size, void* d_ws, size_t ws_size,
                              hipStream_t stream) {
    const float* x  = (const float*)d_in[0];  // [B,C,H,W]
    const float* w1 = (const float*)d_in[1];  // [HIDDEN,C]
    const float* w2 = (const float*)d_in[2];  // [C,HIDDEN,Q]
    const float* wf = (const float*)d_in[3];  // [C,2C]
    float* out = (float*)d_out;               // [B,C,H,W]

    float* ws    = (float*)d_ws;
    float* flat  = ws;                                  // B*C*N      = 786432
    float* qvals = flat  + (size_t)B_ * C_ * N_;        // B*C*Q      = 16384
    float* scale = qvals + (size_t)B_ * C_ * Q_;        // B*C        = 4096
    float* weff  = scale + (size_t)B_ * C_;             // B*C*C      = 1048576

    k_avgpool  <<<(B_ * C_ * N_ + 255) / 256, 256, 0, stream>>>(x, flat);
    k_softquant<<<B_ * C_,                    256, 0, stream>>>(flat, qvals);
    k_mlp_scale<<<B_,                         256, 0, stream>>>(qvals, w1, w2, scale);
    k_weff     <<<(B_ * C_ * C_) / 256,       256, 0, stream>>>(wf, scale, weff);
    k_fuse_wmma<<<B_ * (HW_ / NB),            512, 0, stream>>>(x, weff, out);
}